// TokenReorderer_54219667145007
// MI455X (gfx1250) — compile-verified
//
#include <hip/hip_runtime.h>
#include <hip/hip_bf16.h>
#include <stdint.h>

// Tunables ------------------------------------------------------------------
#define THREADS  256          // 8 wave32 waves per block
#define WAVES    (THREADS / 32)
#define EPT      8            // elements per thread
#define CHUNK    (THREADS * EPT)   // 2048 assignments per block
#define NEXPERTS 64

static_assert(CHUNK == THREADS * EPT, "chunk mismatch");

// --- wave32 ballot ---------------------------------------------------------
__device__ __forceinline__ unsigned ballot32(bool p) {
#if __has_builtin(__builtin_amdgcn_ballot_w32)
    return __builtin_amdgcn_ballot_w32(p);
#else
    return (unsigned)__ballot(p);
#endif
}

// mask of lanes holding the same expert id (6 ballots = log2(64 experts))
__device__ __forceinline__ unsigned expert_mask(bool valid, int em) {
    unsigned mask = ballot32(valid);
#pragma unroll
    for (int b = 0; b < 6; ++b) {
        bool bit = (em >> b) & 1;
        unsigned bal = ballot32(bit);
        mask &= bit ? bal : ~bal;
    }
    return mask;
}

// --- gfx1250 async global->LDS staging -------------------------------------
#if __has_builtin(__builtin_amdgcn_global_load_async_to_lds_b128) && \
    __has_builtin(__builtin_amdgcn_s_wait_asynccnt)
#define HAVE_ASYNC_LDS 1
#else
#define HAVE_ASYNC_LDS 0
#endif

typedef int v4i __attribute__((vector_size(16)));
// builtin signature: (v4i addrspace(1)*, v4i addrspace(3)*, imm offset, imm cpol)
#define GLOBAL_V4I(p) ((__attribute__((address_space(1))) v4i*)(p))
#define LDS_V4I(p)    ((__attribute__((address_space(3))) v4i*)(p))

// ===========================================================================
// Kernel 1: per-block expert histogram (LDS atomics), write to workspace.
// Full-chunk fast path: two b128 loads per thread, one wait, then atomics.
// ===========================================================================
__global__ void __launch_bounds__(THREADS)
tr_hist(const int* __restrict__ experts, int M, unsigned* __restrict__ blockHist) {
    __shared__ unsigned hist[NEXPERTS];
    const int t = threadIdx.x;
    if (t < NEXPERTS) hist[t] = 0u;
    __syncthreads();

    const int base = blockIdx.x * CHUNK;
    if (base + CHUNK <= M) {
        const int4* vp = (const int4*)(experts + base) + (t * 2);
        int4 a = vp[0];                       // elements t*8 .. t*8+3
        int4 b = vp[1];                       // elements t*8+4 .. t*8+7
        atomicAdd(&hist[a.x & (NEXPERTS - 1)], 1u);
        atomicAdd(&hist[a.y & (NEXPERTS - 1)], 1u);
        atomicAdd(&hist[a.z & (NEXPERTS - 1)], 1u);
        atomicAdd(&hist[a.w & (NEXPERTS - 1)], 1u);
        atomicAdd(&hist[b.x & (NEXPERTS - 1)], 1u);
        atomicAdd(&hist[b.y & (NEXPERTS - 1)], 1u);
        atomicAdd(&hist[b.z & (NEXPERTS - 1)], 1u);
        atomicAdd(&hist[b.w & (NEXPERTS - 1)], 1u);
    } else {
        for (int v = 0; v < EPT; ++v) {
            int i = base + v * THREADS + t;
            if (i < M) atomicAdd(&hist[experts[i] & (NEXPERTS - 1)], 1u);
        }
    }
    __syncthreads();
    if (t < NEXPERTS) blockHist[blockIdx.x * NEXPERTS + t] = hist[t];
}

// ===========================================================================
// Kernel 2: one block. Per-expert totals -> counts output, exclusive prefix
// over experts, then convert blockHist in-place into global start offsets.
// ===========================================================================
__global__ void __launch_bounds__(NEXPERTS)
tr_scan(unsigned* __restrict__ blockHist, int numBlocks, float* __restrict__ countsOut) {
    __shared__ unsigned totals[NEXPERTS];
    const int e = threadIdx.x;   // one thread per expert

    unsigned tot = 0u;
    for (int b = 0; b < numBlocks; ++b) tot += blockHist[b * NEXPERTS + e];
    totals[e]    = tot;
    countsOut[e] = (float)tot;
    __syncthreads();

    unsigned off = 0u;
    for (int j = 0; j < e; ++j) off += totals[j];      // exclusive prefix (E=64, cheap)

    for (int b = 0; b < numBlocks; ++b) {              // column scan over blocks
        unsigned c = blockHist[b * NEXPERTS + e];
        blockHist[b * NEXPERTS + e] = off;
        off += c;
    }
}

// ===========================================================================
// Kernel 3: stable scatter. Stage chunk to LDS (async copy), build per-wave
// hists (ballot-aggregated), scan vs block offsets, replay with stable ranks.
// ===========================================================================
__global__ void __launch_bounds__(THREADS)
tr_scatter(const float* __restrict__ scores, const int* __restrict__ experts,
           const unsigned* __restrict__ blockOff, const int* __restrict__ topk,
           int M, float* __restrict__ scoresOut, float* __restrict__ idxOut) {
    __shared__ alignas(16) int      ldsExpert[CHUNK];
    __shared__ alignas(16) float    ldsScore[CHUNK];
    __shared__ unsigned             waveOff[WAVES * NEXPERTS];

    const int  t    = threadIdx.x;
    const int  wave = t >> 5;
    const int  lane = t & 31;
    const int  base = blockIdx.x * CHUNK;
    const bool full = (base + CHUNK) <= M;
    const int  K    = topk[0];                // scalar broadcast (SMEM load)

    // ---- stage chunk into LDS: one global pass feeds both phases ----
    if (full) {
#if HAVE_ASYNC_LDS
        const int i0 = t * EPT;               // each wave stages exactly its sub-chunk
        __builtin_amdgcn_global_load_async_to_lds_b128(
            GLOBAL_V4I(experts + base + i0),     LDS_V4I(ldsExpert + i0),     0, 0);
        __builtin_amdgcn_global_load_async_to_lds_b128(
            GLOBAL_V4I(experts + base + i0 + 4), LDS_V4I(ldsExpert + i0 + 4), 0, 0);
        __builtin_amdgcn_global_load_async_to_lds_b128(
            GLOBAL_V4I(scores + base + i0),      LDS_V4I(ldsScore + i0),      0, 0);
        __builtin_amdgcn_global_load_async_to_lds_b128(
            GLOBAL_V4I(scores + base + i0 + 4),  LDS_V4I(ldsScore + i0 + 4),  0, 0);
#else
#pragma unroll
        for (int v = 0; v < EPT; ++v) {
            int i = v * THREADS + t;
            ldsExpert[i] = experts[base + i] & (NEXPERTS - 1);
            ldsScore[i]  = scores[base + i];
        }
#endif
    } else {
        for (int v = 0; v < EPT; ++v) {
            int i  = v * THREADS + t;
            int gi = base + i;
            ldsExpert[i] = (gi < M) ? (experts[gi] & (NEXPERTS - 1)) : -1;
            ldsScore[i]  = (gi < M) ? scores[gi] : 0.0f;
        }
    }

    // zero per-wave histograms (disjoint LDS region from staging)
    waveOff[t]           = 0u;
    waveOff[t + THREADS] = 0u;

#if HAVE_ASYNC_LDS
    if (full) __builtin_amdgcn_s_wait_asynccnt(0);
#endif
    __syncthreads();

    unsigned* myHist = &waveOff[wave * NEXPERTS];
    const unsigned ltmask = (1u << lane) - 1u;

    // ---- phase 1: per-wave histogram, one LDS atomic per distinct expert ----
#pragma unroll
    for (int g = 0; g < EPT; ++g) {
        int i = wave * (CHUNK / WAVES) + g * 32 + lane;
        int e = ldsExpert[i];
        const bool valid = (e >= 0);
        const int  em    = e & (NEXPERTS - 1);
        unsigned mask = expert_mask(valid, em);
        if (valid && __popc(mask & ltmask) == 0)      // group leader per expert
            atomicAdd(&myHist[em], (unsigned)__popc(mask));
    }
    __syncthreads();

    // ---- phase 2: turn per-wave counts into running global start offsets ----
    if (t < NEXPERTS) {
        unsigned off = blockOff[blockIdx.x * NEXPERTS + t];
        for (int w = 0; w < WAVES; ++w) {
            unsigned c = waveOff[w * NEXPERTS + t];
            waveOff[w * NEXPERTS + t] = off;
            off += c;
        }
    }
    __syncthreads();

    // ---- phase 3: stable replay, 32 elements per wave-step ----
#pragma unroll
    for (int g = 0; g < EPT; ++g) {
        int   i = wave * (CHUNK / WAVES) + g * 32 + lane;
        int   e = ldsExpert[i];
        float s = ldsScore[i];
        const bool valid = (e >= 0);
        const int  em    = e & (NEXPERTS - 1);

        unsigned mask = expert_mask(valid, em);
        const unsigned rank   = __popc(mask & ltmask);
        const unsigned cnt    = __popc(mask);
        const int      leader = mask ? (__ffs(mask) - 1) : 0;

        unsigned grpBase = 0u;
        if (valid && rank == 0)                 // per-expert leader bumps counter
            grpBase = atomicAdd(&myHist[em], cnt);
        grpBase = __shfl(grpBase, leader, 32);  // broadcast old value to the group

        if (valid) {
            unsigned pos = grpBase + rank;
            scoresOut[pos] = s;
            idxOut[pos]    = (float)((unsigned)(base + i) / (unsigned)K);
        }
    }
}

// ===========================================================================
extern "C" void kernel_launch(void* const* d_in, const int* in_sizes, int n_in,
                              void* d_out, int out_size, void* d_ws, size_t ws_size,
                              hipStream_t stream) {
    (void)n_in; (void)out_size; (void)ws_size;

    const float* scores  = (const float*)d_in[0];   // top_scores (N,K) f32
    const int*   experts = (const int*)d_in[1];     // selected_experts_indices (N,K) i32
    const int*   topk    = (const int*)d_in[3];     // top_k scalar (device)
    const int    M       = in_sizes[0];             // N*K flattened assignments

    float* out       = (float*)d_out;
    float* scoresOut = out;             // [M]  expert-sorted scores
    float* idxOut    = out + M;         // [M]  token index per sorted slot (as f32)
    float* countsOut = out + 2 * M;     // [64] tokens per expert (f32)

    unsigned* blockHist = (unsigned*)d_ws;          // numBlocks x 64 u32
    const int numBlocks = (M + CHUNK - 1) / CHUNK;  // 256 for M=524288

    tr_hist   <<<numBlocks, THREADS,  0, stream>>>(experts, M, blockHist);
    tr_scan   <<<1,         NEXPERTS, 0, stream>>>(blockHist, numBlocks, countsOut);
    tr_scatter<<<numBlocks, THREADS,  0, stream>>>(scores, experts, blockHist, topk,
                                                   M, scoresOut, idxOut);
}